// LambdaLayer2D_4698694222461
// MI455X (gfx1250) — compile-verified
//
#include <hip/hip_runtime.h>
#include <hip/hip_bf16.h>

typedef __attribute__((ext_vector_type(2))) float v2f;
typedef __attribute__((ext_vector_type(8))) float v8f;

// ---------------------------------------------------------------------------
// Problem constants
// ---------------------------------------------------------------------------
#define BATCH   16
#define CIN     256
#define HH      64
#define WW      64
#define NPIX    4096        // HH*WW
#define HEADS   4
#define DKK     16
#define DVV     64
#define RR      23
#define PADR    11
#define KSQ     529         // 23*23
#define KPAD    533         // padded K row stride (odd mod 64 -> conflict-free)

__device__ __forceinline__ v8f wmma4(v2f a, v2f b, v8f c) {
    // D = A(16x4,f32) * B(4x16,f32) + C(16x16,f32)   (wave32)
    return __builtin_amdgcn_wmma_f32_16x16x4_f32(
        /*neg_a=*/false, a, /*neg_b=*/false, b,
        /*c_mod=*/(short)0, c, /*reuse_a=*/false, /*reuse_b=*/false);
}

// ---------------------------------------------------------------------------
// Kernel 1: q/k/v 1x1 convs as GEMM  W(144x256) @ x[b](256x4096)
// 9 waves per block; wave w owns 16 output channels; block owns a 64-wide n tile.
// ---------------------------------------------------------------------------
__global__ __launch_bounds__(288)
void qkv_gemm(const float* __restrict__ x,
              const float* __restrict__ Wq, const float* __restrict__ Wk,
              const float* __restrict__ Wv,
              float* __restrict__ q_raw, float* __restrict__ k_raw,
              float* __restrict__ v_raw) {
    const int blk  = blockIdx.x;            // 16 batches * 64 tiles
    const int b    = blk >> 6;
    const int n0   = (blk & 63) * 64;
    const int wave = threadIdx.x >> 5;      // 0..8  -> M tile
    const int lane = threadIdx.x & 31;
    const int half = lane >> 4;
    const int l15  = lane & 15;

    const int m0 = wave * 16;
    const float* Wsel;
    float* dst;
    if (m0 < 64) {
        Wsel = Wq + (size_t)m0 * CIN;
        dst  = q_raw + ((size_t)b * 64 + m0) * NPIX + n0;
    } else if (m0 < 80) {
        Wsel = Wk + (size_t)(m0 - 64) * CIN;
        dst  = k_raw + ((size_t)b * 16 + (m0 - 64)) * NPIX + n0;
    } else {
        Wsel = Wv + (size_t)(m0 - 80) * CIN;
        dst  = v_raw + ((size_t)b * 64 + (m0 - 80)) * NPIX + n0;
    }
    const float* xb = x + (size_t)b * CIN * NPIX + n0;

    v8f acc0 = {}, acc1 = {}, acc2 = {}, acc3 = {};
    for (int k0 = 0; k0 < CIN; k0 += 4) {
        const int ka = k0 + 2 * half;
        v2f a;
        a.x = Wsel[l15 * CIN + ka];
        a.y = Wsel[l15 * CIN + ka + 1];
        const float* xk0 = xb + (size_t)ka * NPIX;
        const float* xk1 = xk0 + NPIX;
        v2f b0, b1, b2, b3;
        b0.x = xk0[l15];        b0.y = xk1[l15];
        b1.x = xk0[16 + l15];   b1.y = xk1[16 + l15];
        b2.x = xk0[32 + l15];   b2.y = xk1[32 + l15];
        b3.x = xk0[48 + l15];   b3.y = xk1[48 + l15];
        acc0 = wmma4(a, b0, acc0);
        acc1 = wmma4(a, b1, acc1);
        acc2 = wmma4(a, b2, acc2);
        acc3 = wmma4(a, b3, acc3);
    }
#pragma unroll
    for (int i = 0; i < 8; ++i) {
        const int row = i + 8 * half;
        float* dp = dst + (size_t)row * NPIX;
        dp[l15]      = acc0[i];
        dp[16 + l15] = acc1[i];
        dp[32 + l15] = acc2[i];
        dp[48 + l15] = acc3[i];
    }
}

// ---------------------------------------------------------------------------
// Kernel 2: BatchNorm batch stats (biased) -> fused scale/shift per channel.
// 128 blocks: 0..63 q channels, 64..127 v channels. Deterministic tree reduce.
// ---------------------------------------------------------------------------
__global__ __launch_bounds__(256)
void bn_stats(const float* __restrict__ q_raw, const float* __restrict__ v_raw,
              const float* __restrict__ gq, const float* __restrict__ bq,
              const float* __restrict__ gv, const float* __restrict__ bv,
              float* __restrict__ qscale, float* __restrict__ qshift,
              float* __restrict__ vscale, float* __restrict__ vshift) {
    const int  chg  = blockIdx.x;
    const bool is_v = chg >= 64;
    const int  ch   = is_v ? chg - 64 : chg;
    const float* src = (is_v ? v_raw : q_raw) + (size_t)ch * NPIX;

    float s = 0.f, ss = 0.f;
    for (int i = threadIdx.x; i < BATCH * NPIX; i += 256) {
        const int bb = i >> 12, n = i & (NPIX - 1);
        const float v = src[(size_t)bb * 64 * NPIX + n];
        s += v; ss += v * v;
    }
    __shared__ float S[256], SS[256];
    S[threadIdx.x] = s; SS[threadIdx.x] = ss;
    __syncthreads();
    for (int off = 128; off > 0; off >>= 1) {
        if (threadIdx.x < off) {
            S[threadIdx.x]  += S[threadIdx.x + off];
            SS[threadIdx.x] += SS[threadIdx.x + off];
        }
        __syncthreads();
    }
    if (threadIdx.x == 0) {
        const float inv  = 1.0f / (float)(BATCH * NPIX);
        const float mean = S[0] * inv;
        const float var  = SS[0] * inv - mean * mean;
        const float g    = is_v ? gv[ch] : gq[ch];
        const float be   = is_v ? bv[ch] : bq[ch];
        const float sc   = g * rsqrtf(var + 1e-5f);
        if (is_v) { vscale[ch] = sc; vshift[ch] = be - mean * sc; }
        else      { qscale[ch] = sc; qshift[ch] = be - mean * sc; }
    }
}

// ---------------------------------------------------------------------------
// Kernel 3: softmax row stats for k: max and sum(exp(x-max)) per (b,dk).
// ---------------------------------------------------------------------------
__global__ __launch_bounds__(256)
void softmax_stats(const float* __restrict__ k_raw,
                   float* __restrict__ kmax, float* __restrict__ ksum) {
    const int row = blockIdx.x;                 // b*16 + dk
    const float* src = k_raw + (size_t)row * NPIX;
    __shared__ float S[256];
    __shared__ float rowmax;

    float m = -3.402823e38f;
    for (int i = threadIdx.x; i < NPIX; i += 256) m = fmaxf(m, src[i]);
    S[threadIdx.x] = m; __syncthreads();
    for (int off = 128; off > 0; off >>= 1) {
        if (threadIdx.x < off) S[threadIdx.x] = fmaxf(S[threadIdx.x], S[threadIdx.x + off]);
        __syncthreads();
    }
    if (threadIdx.x == 0) rowmax = S[0];
    __syncthreads();

    float s = 0.f;
    for (int i = threadIdx.x; i < NPIX; i += 256) s += expf(src[i] - rowmax);
    S[threadIdx.x] = s; __syncthreads();
    for (int off = 128; off > 0; off >>= 1) {
        if (threadIdx.x < off) S[threadIdx.x] += S[threadIdx.x + off];
        __syncthreads();
    }
    if (threadIdx.x == 0) { kmax[row] = rowmax; ksum[row] = S[0]; }
}

// ---------------------------------------------------------------------------
// Kernel 4: lambda_c[b] = softmax(k)[16x4096] @ v_bn^T[4096x64] via WMMA.
// 8 waves split the m dimension; LDS combine. One block per batch.
// ---------------------------------------------------------------------------
__global__ __launch_bounds__(256)
void lambda_c_kernel(const float* __restrict__ k_raw, const float* __restrict__ v_raw,
                     const float* __restrict__ kmax, const float* __restrict__ ksum,
                     const float* __restrict__ vscale, const float* __restrict__ vshift,
                     float* __restrict__ lambda_c) {
    const int b    = blockIdx.x;
    const int wave = threadIdx.x >> 5;
    const int lane = threadIdx.x & 31;
    const int half = lane >> 4;
    const int l15  = lane & 15;

    const float* kb = k_raw + (size_t)b * 16 * NPIX;
    const float* vb = v_raw + (size_t)b * 64 * NPIX;
    const float rmax = kmax[b * 16 + l15];
    const float rinv = 1.0f / ksum[b * 16 + l15];

    v8f acc[4] = {{}, {}, {}, {}};
    const int mstart = wave * 512;
    for (int m0 = mstart; m0 < mstart + 512; m0 += 4) {
        const int ma = m0 + 2 * half;
        v2f a;
        a.x = expf(kb[(size_t)l15 * NPIX + ma]     - rmax) * rinv;
        a.y = expf(kb[(size_t)l15 * NPIX + ma + 1] - rmax) * rinv;
#pragma unroll
        for (int t = 0; t < 4; ++t) {
            const int vch = t * 16 + l15;
            const float sc = vscale[vch], sh = vshift[vch];
            v2f bf;
            bf.x = vb[(size_t)vch * NPIX + ma]     * sc + sh;
            bf.y = vb[(size_t)vch * NPIX + ma + 1] * sc + sh;
            acc[t] = wmma4(a, bf, acc[t]);
        }
    }

    __shared__ float red[8][1024];
#pragma unroll
    for (int t = 0; t < 4; ++t)
#pragma unroll
        for (int i = 0; i < 8; ++i)
            red[wave][(i + 8 * half) * 64 + t * 16 + l15] = acc[t][i];
    __syncthreads();
    for (int e = threadIdx.x; e < 1024; e += 256) {
        float s = 0.f;
#pragma unroll
        for (int w2 = 0; w2 < 8; ++w2) s += red[w2][e];
        lambda_c[(size_t)b * 1024 + e] = s;
    }
}

// ---------------------------------------------------------------------------
// Kernel 5: fused 23x23 positional conv (implicit GEMM on WMMA) + q contraction.
// Block = 8 waves handles one (b, vch, 2-row strip).
// LDS stages both the 24x86 image halo AND the 16x529 pos_w matrix (zero-padded
// to K=532 so the K-remainder is branch-free). Accumulator seeded with
// lambda_c + pos_b so y_c + y_p come out together.
// ---------------------------------------------------------------------------
__global__ __launch_bounds__(256)
void fused_pos(const float* __restrict__ q_raw, const float* __restrict__ v_raw,
               const float* __restrict__ qscale, const float* __restrict__ qshift,
               const float* __restrict__ vscale, const float* __restrict__ vshift,
               const float* __restrict__ lambda_c,
               const float* __restrict__ pos_w, const float* __restrict__ pos_b,
               float* __restrict__ out) {
    const int ypair = blockIdx.x;   // 0..31
    const int vch   = blockIdx.y;   // 0..63
    const int b     = blockIdx.z;   // 0..15
    const int wave  = threadIdx.x >> 5;
    const int lane  = threadIdx.x & 31;
    const int half  = lane >> 4;
    const int l15   = lane & 15;

    const int y0 = ypair * 2;
    const float vsc = vscale[vch], vsh = vshift[vch];
    const float* vb = v_raw + ((size_t)b * 64 + vch) * NPIX;

    __shared__ float s_img[24][88];       // rows y0-11..y0+12, cols -11..74
    __shared__ float s_w[DKK][KPAD];      // pos_w, zero-padded cols 529..532

    // Stage pos_w (zero-pad K) into LDS
    for (int idx = threadIdx.x; idx < DKK * KPAD; idx += 256) {
        const int k = idx / KPAD, rs = idx - k * KPAD;
        s_w[k][rs] = (rs < KSQ) ? pos_w[k * KSQ + rs] : 0.f;
    }
    // Stage BN'd image halo (zero border) into LDS
    for (int idx = threadIdx.x; idx < 24 * 86; idx += 256) {
        const int r = idx / 86, c = idx - r * 86;
        const int gy = y0 - PADR + r, gx = c - PADR;
        float val = 0.f;
        if (gy >= 0 && gy < HH && gx >= 0 && gx < WW)
            val = vb[gy * WW + gx] * vsc + vsh;
        s_img[r][c] = val;
    }
    __syncthreads();

    const int ty = wave >> 2;           // 0/1 row within strip
    const int x0 = (wave & 3) * 16;     // 16-wide position tile

    // Seed accumulator: lambda_c[b,k,vch] + pos_b[k]
    v8f acc;
#pragma unroll
    for (int i = 0; i < 8; ++i) {
        const int k = i + 8 * half;
        acc[i] = lambda_c[(size_t)b * 1024 + k * 64 + vch] + pos_b[k];
    }

    // Implicit GEMM: A = pos_w (16 x 529), B = patches (529 x 16)
    for (int rs0 = 0; rs0 < 528; rs0 += 4) {
        const int ka = rs0 + 2 * half;
        v2f a, bf;
        a.x = s_w[l15][ka];
        a.y = s_w[l15][ka + 1];
        const int r0 = ka / RR,       s0 = ka - r0 * RR;
        const int r1 = (ka + 1) / RR, s1 = (ka + 1) - r1 * RR;
        bf.x = s_img[ty + r0][x0 + l15 + s0];
        bf.y = s_img[ty + r1][x0 + l15 + s1];
        acc = wmma4(a, bf, acc);
    }
    {   // K remainder (rs 528..531): A cols 529..531 are zero in LDS, so this
        // is branch-free; B reads stay in-bounds and pair only with zero A.
        v2f a, bf;
        a.x  = s_w[l15][528 + 2 * half];
        a.y  = s_w[l15][529 + 2 * half];
        bf.x = s_img[ty + 22][x0 + l15 + 22];   // rs=528 -> r=22, s=22
        bf.y = 0.f;
        acc = wmma4(a, bf, acc);
    }

    // Contract with BN'd q over k and write Y[b, h*64+vch, n]
    const int n = (y0 + ty) * WW + x0 + l15;
    const float* qb = q_raw + (size_t)b * 64 * NPIX;
    float oh[HEADS] = {0.f, 0.f, 0.f, 0.f};
#pragma unroll
    for (int i = 0; i < 8; ++i) {
        const int k = i + 8 * half;
        const float L = acc[i];
#pragma unroll
        for (int h = 0; h < HEADS; ++h) {
            const int qc = h * 16 + k;
            const float qv = qb[(size_t)qc * NPIX + n] * qscale[qc] + qshift[qc];
            oh[h] += qv * L;
        }
    }
#pragma unroll
    for (int h = 0; h < HEADS; ++h) oh[h] += __shfl_xor(oh[h], 16, 32);
    if (half == 0) {
#pragma unroll
        for (int h = 0; h < HEADS; ++h)
            out[((size_t)b * 256 + h * 64 + vch) * NPIX + n] = oh[h];
    }
}

// ---------------------------------------------------------------------------
// Launcher
// ---------------------------------------------------------------------------
extern "C" void kernel_launch(void* const* d_in, const int* in_sizes, int n_in,
                              void* d_out, int out_size, void* d_ws, size_t ws_size,
                              hipStream_t stream) {
    const float* x     = (const float*)d_in[0];
    const float* Wq    = (const float*)d_in[1];
    const float* Wk    = (const float*)d_in[2];
    const float* Wv    = (const float*)d_in[3];
    const float* gq    = (const float*)d_in[4];
    const float* bq    = (const float*)d_in[5];
    const float* gv    = (const float*)d_in[6];
    const float* bv    = (const float*)d_in[7];
    const float* pos_w = (const float*)d_in[8];
    const float* pos_b = (const float*)d_in[9];
    float* out = (float*)d_out;

    float* ws       = (float*)d_ws;
    float* q_raw    = ws;                                   // 16*64*4096
    float* k_raw    = q_raw + (size_t)BATCH * 64 * NPIX;    // 16*16*4096
    float* v_raw    = k_raw + (size_t)BATCH * 16 * NPIX;    // 16*64*4096
    float* qscale   = v_raw + (size_t)BATCH * 64 * NPIX;
    float* qshift   = qscale + 64;
    float* vscale   = qshift + 64;
    float* vshift   = vscale + 64;
    float* kmax     = vshift + 64;                          // 256
    float* ksum     = kmax + 256;                           // 256
    float* lambda_c = ksum + 256;                           // 16*1024

    qkv_gemm<<<dim3(BATCH * 64), dim3(288), 0, stream>>>(
        x, Wq, Wk, Wv, q_raw, k_raw, v_raw);

    bn_stats<<<dim3(128), dim3(256), 0, stream>>>(
        q_raw, v_raw, gq, bq, gv, bv, qscale, qshift, vscale, vshift);

    softmax_stats<<<dim3(BATCH * 16), dim3(256), 0, stream>>>(k_raw, kmax, ksum);

    lambda_c_kernel<<<dim3(BATCH), dim3(256), 0, stream>>>(
        k_raw, v_raw, kmax, ksum, vscale, vshift, lambda_c);

    fused_pos<<<dim3(32, 64, BATCH), dim3(256), 0, stream>>>(
        q_raw, v_raw, qscale, qshift, vscale, vshift, lambda_c, pos_w, pos_b, out);
}